// BimodalClassifier_1176821039550
// MI455X (gfx1250) — compile-verified
//
#include <hip/hip_runtime.h>
#include <hip/hip_bf16.h>
#include <math.h>

#define BB 8
#define NN 16384
#define MM 64
#define EE 49152
#define CIN 128
#define CC 128            // C_OUT
#define NEG_SLOPE 0.2f
#define MARGIN 4.2f

typedef __attribute__((ext_vector_type(2))) float v2f;
typedef __attribute__((ext_vector_type(8))) float v8f;

// ---------------- workspace layout (in floats) ----------------
#define OFF_XP   ((size_t)0)                         // [B*N, C] projected x
#define OFF_ES   (OFF_XP  + (size_t)BB*NN*CC)        // [M,B,C] edge_sums
#define OFF_XE   (OFF_ES  + (size_t)MM*BB*CC)        // [M,B,C] x_edge
#define OFF_D    (OFF_XE  + (size_t)MM*BB*CC)        // [N] node degree
#define OFF_ED   (OFF_D   + (size_t)NN)              // [M] edge degree
#define OFF_BN   (OFF_ED  + (size_t)MM)              // [M] 1/deg
#define OFF_NM   (OFF_BN  + (size_t)MM)              // [N*B] node max (ordered uint)
#define OFF_NS   (OFF_NM  + (size_t)NN*BB)           // [N*B] node exp-sum
#define OFF_SC   (OFF_NS  + (size_t)NN*BB)           // [E*B] leaky-relu scores
#define OFF_EX   (OFF_SC  + (size_t)EE*BB)           // [E*B] exp(score - max)
#define OFF_NRM  (OFF_EX  + (size_t)EE*BB)           // [M] edge_flat norms
#define OFF_RS   (OFF_NRM + (size_t)MM)              // [M] edge_flat row sums
#define OFF_ACC  (OFF_RS  + (size_t)MM)              // [8] scalar accumulators
#define WS_FLOATS (OFF_ACC + 8)

// ordered-uint encoding so unsigned atomicMax == float max
__device__ __forceinline__ unsigned f2ord(float f) {
  unsigned u = __float_as_uint(f);
  return (u & 0x80000000u) ? ~u : (u | 0x80000000u);
}
__device__ __forceinline__ float ord2f(unsigned u) {
  return (u & 0x80000000u) ? __uint_as_float(u ^ 0x80000000u) : __uint_as_float(~u);
}

// ---------------------------------------------------------------
__global__ void zero_kernel(float* __restrict__ p, size_t n) {
  size_t i = (size_t)blockIdx.x * blockDim.x + threadIdx.x;
  size_t s = (size_t)gridDim.x * blockDim.x;
  for (; i < n; i += s) p[i] = 0.0f;
}

// ---------------- GEMM: xp[B*N, C] = x[B*N, CIN] @ w[CIN, C] ----
// wave32 WMMA f32 16x16x4; weight staged in LDS (64 KB of 320 KB/WGP).
__global__ void gemm_wmma_kernel(const float* __restrict__ x,
                                 const float* __restrict__ w,
                                 float* __restrict__ xp, int total_rows) {
  __shared__ float lw[CIN * CC];       // 64 KB
  const int tid = threadIdx.x;         // 256 threads = 8 wave32
  {
    const float4* w4 = (const float4*)w;
    float4* l4 = (float4*)lw;
    for (int i = tid; i < CIN * CC / 4; i += 256) l4[i] = w4[i];
  }
  __syncthreads();

  const int wave    = tid >> 5;        // 0..7 -> output column tile
  const int lane    = tid & 31;
  const int lhalf   = lane >> 4;       // 0/1 selects K pair (per ISA A layout)
  const int lmod    = lane & 15;       // M row / N col within tile
  const int colBase = wave * 16;

  const int ntiles = total_rows >> 4;
  for (int rt = blockIdx.x; rt < ntiles; rt += gridDim.x) {
    const int row0 = rt << 4;
    const float* arow = x + (size_t)(row0 + lmod) * CIN;
    v8f acc = {};
#pragma unroll
    for (int k = 0; k < CIN; k += 4) {
      const int ka = k + 2 * lhalf;
      v2f a = {arow[ka], arow[ka + 1]};
      v2f b = {lw[ka * CC + colBase + lmod], lw[(ka + 1) * CC + colBase + lmod]};
      acc = __builtin_amdgcn_wmma_f32_16x16x4_f32(false, a, false, b,
                                                  (short)0, acc, false, false);
    }
    float* orow = xp + (size_t)row0 * CC + colBase + lmod;
#pragma unroll
    for (int v = 0; v < 8; ++v)
      orow[(size_t)(v + 8 * lhalf) * CC] = acc[v];
  }
}

// ---------------- degrees -------------------------------------
__global__ void degree_kernel(const int* __restrict__ node_index,
                              const int* __restrict__ edge_index,
                              float* __restrict__ D, float* __restrict__ ED) {
  int e = blockIdx.x * blockDim.x + threadIdx.x;
  if (e < EE) {
    atomicAdd(&D[node_index[e]], 1.0f);
    atomicAdd(&ED[edge_index[e]], 1.0f);
  }
}

// ---------------- edge_sums scatter (LDS-staged) + sum(x_i) ----
// grid: (chunks, B); 256 thr; LDS image of [64 x 128] slice = 32 KB.
#define CHUNK_E 1024
__global__ void edge_sums_kernel(const float* __restrict__ xp,
                                 const int* __restrict__ node_index,
                                 const int* __restrict__ edge_index,
                                 float* __restrict__ edge_sums,
                                 float* __restrict__ scalars) {
  __shared__ float ls[MM * CC];        // 32 KB
  __shared__ float red[256];
  const int tid = threadIdx.x;
  const int b   = blockIdx.y;
  const int e0  = blockIdx.x * CHUNK_E;
  for (int i = tid; i < MM * CC; i += 256) ls[i] = 0.0f;
  __syncthreads();

  float sxi = 0.0f;
  const int c  = tid & 127;
  const int el = tid >> 7;             // 0/1 : two edges per pass
  for (int i = 0; i < CHUNK_E; i += 2) {
    const int e = e0 + i + el;
    if (e < EE) {
      const int n = node_index[e];
      const int m = edge_index[e];
      const float v = xp[((size_t)b * NN + n) * CC + c];
      atomicAdd(&ls[m * CC + c], v);
      sxi += v;
    }
  }
  __syncthreads();
  for (int i = tid; i < MM * CC; i += 256) {
    const float v = ls[i];
    if (v != 0.0f) {
      const int m = i >> 7, cc = i & 127;
      atomicAdd(&edge_sums[((size_t)m * BB + b) * CC + cc], v);
    }
  }
  red[tid] = sxi;
  __syncthreads();
  for (int o = 128; o > 0; o >>= 1) {
    if (tid < o) red[tid] += red[tid + o];
    __syncthreads();
  }
  if (tid == 0) atomicAdd(&scalars[0], red[0]);
}

// ---------------- per-hyperedge norms / rowsums / 1/deg --------
__global__ void edge_norm_kernel(const float* __restrict__ edge_sums,
                                 const float* __restrict__ ED,
                                 float* __restrict__ norms,
                                 float* __restrict__ rowsum,
                                 float* __restrict__ Bnorm) {
  __shared__ float rs[256], rq[256];
  const int m = blockIdx.x, tid = threadIdx.x;
  float s = 0.0f, q = 0.0f;
  for (int k = tid; k < BB * CC; k += 256) {
    const float v = edge_sums[(size_t)m * BB * CC + k];
    s += v; q += v * v;
  }
  rs[tid] = s; rq[tid] = q;
  __syncthreads();
  for (int o = 128; o > 0; o >>= 1) {
    if (tid < o) { rs[tid] += rs[tid + o]; rq[tid] += rq[tid + o]; }
    __syncthreads();
  }
  if (tid == 0) {
    norms[m]  = sqrtf(rq[0]);
    rowsum[m] = rs[0];
    const float d = ED[m];
    Bnorm[m] = (d > 0.0f) ? 1.0f / d : 0.0f;
  }
}

// ---------------- pairwise contrastive loss (M=64) -------------
__global__ void hyper_loss_kernel(const float* __restrict__ edge_sums,
                                  const float* __restrict__ norms,
                                  float* __restrict__ scalars) {
  __shared__ float rd[256], rq[256];
  const int i = blockIdx.x, tid = threadIdx.x;
  const float ni = fmaxf(norms[i], 1e-12f);
  float acc = 0.0f;
  for (int j = 0; j < MM; ++j) {
    float pd = 0.0f, ps = 0.0f;
    for (int k = tid; k < BB * CC; k += 256) {
      const float a = edge_sums[(size_t)i * BB * CC + k];
      const float c = edge_sums[(size_t)j * BB * CC + k];
      pd += a * c;
      const float df = a - c;
      ps += df * df;
    }
    rd[tid] = pd; rq[tid] = ps;
    __syncthreads();
    for (int o = 128; o > 0; o >>= 1) {
      if (tid < o) { rd[tid] += rd[tid + o]; rq[tid] += rq[tid + o]; }
      __syncthreads();
    }
    if (tid == 0) {
      const float nj  = fmaxf(norms[j], 1e-12f);
      const float cos = rd[0] / (ni * nj);
      const float dst = sqrtf(rq[0] + 1e-12f);
      acc += cos * dst + (1.0f - cos) * fmaxf(MARGIN - dst, 0.0f);
    }
    __syncthreads();
  }
  if (tid == 0) atomicAdd(&scalars[1], acc);
}

// ---------------- attention scores + segment max ---------------
// one block per edge; wave w handles batch b = w (8 wave32).
__global__ void score_kernel(const float* __restrict__ xp,
                             const float* __restrict__ edge_sums,
                             const float* __restrict__ att,
                             const int* __restrict__ node_index,
                             const int* __restrict__ edge_index,
                             float* __restrict__ escore,
                             unsigned* __restrict__ node_max) {
  const int e = blockIdx.x;
  const int b = threadIdx.x >> 5;
  const int lane = threadIdx.x & 31;
  const int n = node_index[e];
  const int m = edge_index[e];
  const float4 xi = ((const float4*)(xp + ((size_t)b * NN + n) * CC))[lane];
  const float4 xj = ((const float4*)(edge_sums + ((size_t)m * BB + b) * CC))[lane];
  const float4 aa = ((const float4*)att)[lane];
  const float4 ab = ((const float4*)(att + CC))[lane];
  float p = xi.x * aa.x + xi.y * aa.y + xi.z * aa.z + xi.w * aa.w
          + xj.x * ab.x + xj.y * ab.y + xj.z * ab.z + xj.w * ab.w;
#pragma unroll
  for (int o = 16; o > 0; o >>= 1) p += __shfl_xor(p, o, 32);
  if (lane == 0) {
    const float s = (p >= 0.0f) ? p : NEG_SLOPE * p;
    escore[(size_t)e * BB + b] = s;
    atomicMax(&node_max[(size_t)n * BB + b], f2ord(s));
  }
}

// ---------------- exp(score - segmax) + segment sum ------------
__global__ void expsum_kernel(const float* __restrict__ escore,
                              const unsigned* __restrict__ node_max,
                              const int* __restrict__ node_index,
                              float* __restrict__ exv,
                              float* __restrict__ node_sum) {
  size_t i = (size_t)blockIdx.x * blockDim.x + threadIdx.x;
  const size_t tot = (size_t)EE * BB;
  const size_t s = (size_t)gridDim.x * blockDim.x;
  for (; i < tot; i += s) {
    const int e = (int)(i >> 3), b = (int)(i & 7);
    const int n = node_index[e];
    const float mx = ord2f(node_max[(size_t)n * BB + b]);
    const float ex = __expf(escore[i] - mx);
    exv[i] = ex;
    atomicAdd(&node_sum[(size_t)n * BB + b], ex);
  }
}

// ---------------- node->edge scatter (LDS-staged) --------------
__global__ void n2e_kernel(const float* __restrict__ xp,
                           const float* __restrict__ exv,
                           const float* __restrict__ node_sum,
                           const float* __restrict__ Bnorm,
                           const int* __restrict__ node_index,
                           const int* __restrict__ edge_index,
                           float* __restrict__ x_edge) {
  __shared__ float ls[MM * CC];        // 32 KB
  const int tid = threadIdx.x;
  const int b   = blockIdx.y;
  const int e0  = blockIdx.x * CHUNK_E;
  for (int i = tid; i < MM * CC; i += 256) ls[i] = 0.0f;
  __syncthreads();
  const int c  = tid & 127;
  const int el = tid >> 7;
  for (int i = 0; i < CHUNK_E; i += 2) {
    const int e = e0 + i + el;
    if (e < EE) {
      const int n = node_index[e];
      const int m = edge_index[e];
      const float alpha = exv[(size_t)e * BB + b] /
                          (node_sum[(size_t)n * BB + b] + 1e-16f);
      const float v = Bnorm[m] * alpha * xp[((size_t)b * NN + n) * CC + c];
      atomicAdd(&ls[m * CC + c], v);
    }
  }
  __syncthreads();
  for (int i = tid; i < MM * CC; i += 256) {
    const float v = ls[i];
    if (v != 0.0f) {
      const int m = i >> 7, cc = i & 127;
      atomicAdd(&x_edge[((size_t)m * BB + b) * CC + cc], v);
    }
  }
}

// ---------------- edge->node scatter into d_out ----------------
__global__ void e2n_kernel(const float* __restrict__ x_edge,
                           const float* __restrict__ exv,
                           const float* __restrict__ node_sum,
                           const float* __restrict__ Dv,
                           const int* __restrict__ node_index,
                           const int* __restrict__ edge_index,
                           float* __restrict__ out) {
  const int tid = threadIdx.x;
  const int e = blockIdx.x * 2 + (tid >> 7);
  const int b = blockIdx.y;
  const int c = tid & 127;
  if (e < EE) {
    const int n = node_index[e];
    const int m = edge_index[e];
    const float alpha = exv[(size_t)e * BB + b] /
                        (node_sum[(size_t)n * BB + b] + 1e-16f);
    const float v = Dv[n] * alpha * x_edge[((size_t)m * BB + b) * CC + c];
    atomicAdd(&out[((size_t)b * NN + n) * CC + c], v);
  }
}

// ---------------- scalar finalize ------------------------------
__global__ void finalize_kernel(const float* __restrict__ ED,
                                const float* __restrict__ rowsum,
                                const float* __restrict__ scalars,
                                float* __restrict__ out_loss) {
  __shared__ float red[64];
  const int t = threadIdx.x;           // 64 threads
  red[t] = ED[t] * rowsum[t];
  __syncthreads();
  for (int o = 32; o > 0; o >>= 1) {
    if (t < o) red[t] += red[t + o];
    __syncthreads();
  }
  if (t == 0) {
    const float sum_xj = red[0];
    const float sum_xi = scalars[0];
    const float hyper  = scalars[1];
    const float denom  = (float)EE * BB * CC;
    const float mean_diff  = (sum_xi - sum_xj) / denom;
    const float loss_hyper = fabsf(hyper / ((float)MM * MM)) /
                             ((float)(MM + 1) * (MM + 1));
    out_loss[0] = fabsf(mean_diff) + loss_hyper;
  }
}

// ---------------------------------------------------------------
extern "C" void kernel_launch(void* const* d_in, const int* in_sizes, int n_in,
                              void* d_out, int out_size, void* d_ws, size_t ws_size,
                              hipStream_t stream) {
  const float* x      = (const float*)d_in[0];
  const float* weight = (const float*)d_in[1];
  const float* att    = (const float*)d_in[2];
  const int* node_index = (const int*)d_in[3];
  const int* edge_index = (const int*)d_in[4];
  float* out = (float*)d_out;

  float* ws = (float*)d_ws;
  float*    xp     = ws + OFF_XP;
  float*    esum   = ws + OFF_ES;
  float*    xedge  = ws + OFF_XE;
  float*    Dv     = ws + OFF_D;
  float*    ED     = ws + OFF_ED;
  float*    Bn     = ws + OFF_BN;
  unsigned* nmax   = (unsigned*)(ws + OFF_NM);
  float*    nsum   = ws + OFF_NS;
  float*    escr   = ws + OFF_SC;
  float*    exv    = ws + OFF_EX;
  float*    norms  = ws + OFF_NRM;
  float*    rowsum = ws + OFF_RS;
  float*    scal   = ws + OFF_ACC;

  // 0) zero init (ordered-uint 0 == minimum encoded float, OK for seg-max)
  zero_kernel<<<2048, 256, 0, stream>>>(ws, WS_FLOATS);
  zero_kernel<<<2048, 256, 0, stream>>>(out, (size_t)BB * NN * CC + 1);

  // 1) projection GEMM via WMMA f32
  gemm_wmma_kernel<<<2048, 256, 0, stream>>>(x, weight, xp, BB * NN);

  // 2) degrees
  degree_kernel<<<(EE + 255) / 256, 256, 0, stream>>>(node_index, edge_index, Dv, ED);

  // 3) edge_sums + sum(x_i)
  {
    dim3 g(EE / CHUNK_E, BB);
    edge_sums_kernel<<<g, 256, 0, stream>>>(xp, node_index, edge_index, esum, scal);
  }

  // 4) norms / rowsums / 1/deg
  edge_norm_kernel<<<MM, 256, 0, stream>>>(esum, ED, norms, rowsum, Bn);

  // 5) pairwise contrastive loss
  hyper_loss_kernel<<<MM, 256, 0, stream>>>(esum, norms, scal);

  // 6) attention scores + per-node max
  score_kernel<<<EE, 256, 0, stream>>>(xp, esum, att, node_index, edge_index,
                                       escr, nmax);

  // 7) exp + segment sums
  expsum_kernel<<<2048, 256, 0, stream>>>(escr, nmax, node_index, exv, nsum);

  // 8) node -> edge
  {
    dim3 g(EE / CHUNK_E, BB);
    n2e_kernel<<<g, 256, 0, stream>>>(xp, exv, nsum, Bn, node_index, edge_index, xedge);
  }

  // 9) edge -> node (accumulate into output)
  {
    dim3 g(EE / 2, BB);
    e2n_kernel<<<g, 256, 0, stream>>>(xedge, exv, nsum, Dv, node_index, edge_index, out);
  }

  // 10) loss scalar
  finalize_kernel<<<1, 64, 0, stream>>>(ED, rowsum, scal, out + (size_t)BB * NN * CC);
}